// GatedDepthwiseConv1d_87230785781944
// MI455X (gfx1250) — compile-verified
//
#include <hip/hip_runtime.h>
#include <hip/hip_bf16.h>
#include <stdint.h>

// Problem constants (match reference setup_inputs):
//   x:       (B=4, S=8192, D=2048) fp32
//   filters: (FO=2, B=4, S=8192, K=3) fp32
#define BB   4
#define SS   8192
#define DD   2048
#define KK   3
#define FO   2

// Phase-2 tiling
#define TS      32             // output t-rows per block
#define RROWS   (TS + 2)       // rows staged in LDS (2-row causal halo)
#define CFLOATS 256            // channel floats per tile
#define CF4     (CFLOATS / 4)  // 64 float4 columns
#define NCT     (DD / CFLOATS) // 8 channel tiles
#define NTT     (SS / TS)      // 256 t tiles
#define THREADS 256

// native clang vector type (required by __builtin_nontemporal_store)
typedef float v4f __attribute__((ext_vector_type(4)));

__device__ __forceinline__ void async_b128_to_lds(unsigned lds_off, const v4f* src) {
    asm volatile("global_load_async_to_lds_b128 %0, %1, off"
                 :: "v"(lds_off), "v"(src)
                 : "memory");
}

// ---------------------------------------------------------------------------
// HOT KERNEL. Streaming 3-tap causal depthwise conv, per-batch scalar taps:
//   out[b,t,c] = w0*x[b,t,c] + w1*x[b,t-1,c] + w2*x[b,t-2,c]
// A (TS+2) x 256-float fp32 tile of x is staged into LDS with CDNA5 async
// global->LDS b128 loads (ASYNCcnt). Halo (t<0) handling is a uniform branch
// so the load loops are branch-free. Waves drain with s_wait_asynccnt, sync,
// then each thread computes 8 float4 outputs from LDS (3 FMAs/elem) and
// streams them out with non-temporal b128 stores.
// ---------------------------------------------------------------------------
__global__ __launch_bounds__(THREADS)
void hyena_conv3_async(const float* __restrict__ x,
                       const float* __restrict__ w,
                       float* __restrict__ out) {
    __shared__ v4f tile[RROWS * CF4];   // 34 * 64 * 16 B = 34 KB

    const int blk = blockIdx.x;
    const int ct  = blk & (NCT - 1);            // channel tile 0..7
    const int tt  = (blk >> 3) & (NTT - 1);     // t tile 0..255
    const int b   = blk >> 11;                  // batch 0..3
    const int tid = threadIdx.x;

    const int t0   = tt * TS;
    const int c0f4 = ct * CF4;                  // channel offset in float4 units

    // per-batch taps (uniform scalars -> s_load_b96)
    const float w0 = w[b * 4 + 0];
    const float w1 = w[b * 4 + 1];
    const float w2 = w[b * 4 + 2];

    // LDS byte offset of tile[] (addrspace(3) offset == low 32 bits of generic)
    const unsigned ldsbase = (unsigned)(uintptr_t)(&tile[0]);

    const v4f* xg = (const v4f*)x;              // x in float4 units; D/4 = 512
    // float4 index of tile element i: row = i>>6 maps to t = t0-2+row
    // global float4 addr = (b*SS + t0 - 2 + row)*512 + c0f4 + col
    const size_t gbase = (size_t)(b * SS + t0 - 2) * (DD / 4) + c0f4;

    if (tt != 0) {
        // all 34 rows valid: branch-free async load loop (9 iters/thread)
        for (int i = tid; i < RROWS * CF4; i += THREADS) {
            const int row = i >> 6;
            const int col = i & (CF4 - 1);
            async_b128_to_lds(ldsbase + (unsigned)i * 16u,
                              xg + (gbase + (size_t)row * (DD / 4) + col));
        }
    } else {
        // rows 0..1 are t<0: zero-fill (uniform), then load rows 2..33
        if (tid < 2 * CF4) tile[tid] = (v4f){0.f, 0.f, 0.f, 0.f};
        for (int i = tid + 2 * CF4; i < RROWS * CF4; i += THREADS) {
            const int row = i >> 6;
            const int col = i & (CF4 - 1);
            async_b128_to_lds(ldsbase + (unsigned)i * 16u,
                              xg + (gbase + (size_t)row * (DD / 4) + col));
        }
    }

    // drain this wave's async queue, then make all waves' LDS writes visible
    asm volatile("s_wait_asynccnt 0" ::: "memory");
    __syncthreads();

    // ---- compute: out[t] = w0*x[t] + w1*x[t-1] + w2*x[t-2] ----
    for (int i = tid; i < TS * CF4; i += THREADS) {
        const int row = i >> 6;       // 0..31
        const int col = i & (CF4 - 1);

        const v4f xm2 = tile[(row + 0) * CF4 + col];  // x[t-2]
        const v4f xm1 = tile[(row + 1) * CF4 + col];  // x[t-1]
        const v4f xc  = tile[(row + 2) * CF4 + col];  // x[t]

        const v4f o = w0 * xc + w1 * xm1 + w2 * xm2;  // contracts to v_fma_f32

        const int t = t0 + row;
        v4f* dst = (v4f*)out + ((size_t)(b * SS + t) * (DD / 4) + c0f4 + col);
        // write-once data: non-temporal store keeps `out` from displacing the
        // x halo lines in L2 (th:TH_STORE_NT)
        __builtin_nontemporal_store(o, dst);
    }
}

// ---------------------------------------------------------------------------
// Phase 1: reduce filters -> per-batch flipped taps, scaled by 1/S.
//   k[b,kk] = (1/S) * sum_{fo,s} filters[fo,b,s,kk]
//   tap j multiplies x[t-j] with k[b, K-1-j]:
//     w[b*4+0] = k[b,2], w[b*4+1] = k[b,1], w[b*4+2] = k[b,0]
// Total input: 768 KB -> runtime is noise vs. phase 2.
// ---------------------------------------------------------------------------
__global__ __launch_bounds__(THREADS)
void hyena_taps_reduce(const float* __restrict__ f, float* __restrict__ w) {
    const int b   = blockIdx.x;   // 0..3
    const int tid = threadIdx.x;  // 0..255

    float s0 = 0.f, s1 = 0.f, s2 = 0.f;
    // iterate over (fo, s) pairs: FO*SS = 16384 triples of K=3 floats
    for (int i = tid; i < FO * SS; i += THREADS) {
        const int fo = i >> 13;          // i / SS
        const int s  = i & (SS - 1);     // i % SS
        const float* p = f + (((size_t)fo * BB + b) * SS + s) * KK;
        s0 += p[0];
        s1 += p[1];
        s2 += p[2];
    }

    __shared__ float r0[THREADS], r1[THREADS], r2[THREADS];
    r0[tid] = s0; r1[tid] = s1; r2[tid] = s2;
    __syncthreads();
    for (int off = THREADS / 2; off > 0; off >>= 1) {
        if (tid < off) {
            r0[tid] += r0[tid + off];
            r1[tid] += r1[tid + off];
            r2[tid] += r2[tid + off];
        }
        __syncthreads();
    }
    if (tid == 0) {
        const float inv = 1.0f / (float)SS;
        w[b * 4 + 0] = r2[0] * inv;  // tap on x[t]   : k[b,2]
        w[b * 4 + 1] = r1[0] * inv;  // tap on x[t-1] : k[b,1]
        w[b * 4 + 2] = r0[0] * inv;  // tap on x[t-2] : k[b,0]
    }
}

// ---------------------------------------------------------------------------
// Launch
// ---------------------------------------------------------------------------
extern "C" void kernel_launch(void* const* d_in, const int* in_sizes, int n_in,
                              void* d_out, int out_size, void* d_ws, size_t ws_size,
                              hipStream_t stream) {
    const float* x = (const float*)d_in[0];   // (4, 8192, 2048) fp32
    const float* f = (const float*)d_in[1];   // (2, 4, 8192, 3) fp32
    float* out = (float*)d_out;
    float* w   = (float*)d_ws;                // 16 floats used

    hyena_taps_reduce<<<BB, THREADS, 0, stream>>>(f, w);

    const int nblocks = BB * NTT * NCT;       // 4 * 256 * 8 = 8192
    hyena_conv3_async<<<nblocks, THREADS, 0, stream>>>(x, w, out);
}